// CoordinateLayer_31078383354577
// MI455X (gfx1250) — compile-verified
//
#include <hip/hip_runtime.h>
#include <hip/hip_bf16.h>
#include <math.h>

typedef __attribute__((ext_vector_type(2))) float v2f;
typedef __attribute__((ext_vector_type(8))) float v8f;

#define NSTEPS 2048
#define BATCH  512
#define NPTS   (NSTEPS * 3)        // 6144 chain steps
#define FRAG_S 16                  // fragment length (one WMMA tile)
#define NFRAG  (NPTS / FRAG_S)     // 384 fragments
#define GROUP  16                  // fragments per scan group
#define NGROUP (NFRAG / GROUP)     // 24 groups

__device__ __forceinline__ void nrm3(float &x, float &y, float &z) {
    float inv = rsqrtf(x * x + y * y + z * z + 1e-12f);
    x *= inv; y *= inv; z *= inv;
}
__device__ __forceinline__ void cross3(float ax, float ay, float az,
                                       float bx, float by, float bz,
                                       float &cx, float &cy, float &cz) {
    cx = ay * bz - az * by;
    cy = az * bx - ax * bz;
    cz = ax * by - ay * bx;
}
// (M,t) <- (M*Q, M*q + t)
__device__ __forceinline__ void compose(float M[9], float t[3],
                                        const float Q[9], const float q[3]) {
    float Mn[9], tn[3];
#pragma unroll
    for (int i = 0; i < 3; ++i) {
#pragma unroll
        for (int j = 0; j < 3; ++j)
            Mn[i * 3 + j] = M[i * 3 + 0] * Q[0 + j] +
                            M[i * 3 + 1] * Q[3 + j] +
                            M[i * 3 + 2] * Q[6 + j];
        tn[i] = M[i * 3 + 0] * q[0] + M[i * 3 + 1] * q[1] +
                M[i * 3 + 2] * q[2] + t[i];
    }
#pragma unroll
    for (int k = 0; k < 9; ++k) M[k] = Mn[k];
    t[0] = tn[0]; t[1] = tn[1]; t[2] = tn[2];
}

// ---------------------------------------------------------------------------
// Stage A: per-(fragment,batch) local NeRF chain from the canonical init frame
// (identity frame), writing local coords to d_out and the fragment end-frame
// affine (M row-major [9], t [3]) to scratch.
// ---------------------------------------------------------------------------
__global__ void nerf_fragment_kernel(const float* __restrict__ dih,
                                     float* __restrict__ out,
                                     float* __restrict__ tw) {
    int tid = blockIdx.x * blockDim.x + threadIdx.x;   // exact grid
    int b = tid % BATCH;
    int f = tid / BATCH;

    const float BL[3] = {145.801f, 152.326f, 132.868f};
    const float BA[3] = {2.124f, 1.941f, 2.028f};
    float rc[3], rs[3];
#pragma unroll
    for (int d = 0; d < 3; ++d) {
        float th = 3.14159265358979323846f - BA[d];
        rc[d] = BL[d] * __cosf(th);
        rs[d] = BL[d] * __sinf(th);
    }

    float ax = -0.70710678f, ay = 1.22474487f, az = 0.0f;
    float bx = -1.41421356f, by = 0.0f,        bz = 0.0f;
    float cx = 0.0f,         cy = 0.0f,        cz = 0.0f;

    int n0 = f * FRAG_S;
    for (int s = 0; s < FRAG_S; ++s) {
        int n = n0 + s;
        int l = n / 3, d = n % 3;
        float phi = dih[((size_t)l * BATCH + b) * 3 + d];
        float sp, cp;
        __sincosf(phi, &sp, &cp);
        float px = rc[d], py = cp * rs[d], pz = sp * rs[d];

        float ux = cx - bx, uy = cy - by, uz = cz - bz;      // bc
        nrm3(ux, uy, uz);
        float vx = bx - ax, vy = by - ay, vz = bz - az;      // b-a
        float nx, ny, nz;
        cross3(vx, vy, vz, ux, uy, uz, nx, ny, nz);          // n
        nrm3(nx, ny, nz);
        float wx, wy, wz;
        cross3(nx, ny, nz, ux, uy, uz, wx, wy, wz);          // n x bc

        float ox = ux * px + wx * py + nx * pz + cx;
        float oy = uy * px + wy * py + ny * pz + cy;
        float oz = uz * px + wz * py + nz * pz + cz;

        float* op = out + ((size_t)n * BATCH + b) * 3;
        op[0] = ox; op[1] = oy; op[2] = oz;

        ax = bx; ay = by; az = bz;
        bx = cx; by = cy; bz = cz;
        cx = ox; cy = oy; cz = oz;
    }

    // end-frame affine from last three local coords
    float ux = cx - bx, uy = cy - by, uz = cz - bz;
    nrm3(ux, uy, uz);
    float vx = bx - ax, vy = by - ay, vz = bz - az;
    float nx, ny, nz;
    cross3(vx, vy, vz, ux, uy, uz, nx, ny, nz);
    nrm3(nx, ny, nz);
    float wx, wy, wz;
    cross3(nx, ny, nz, ux, uy, uz, wx, wy, wz);

    float* T = tw + (size_t)tid * 12;   // [f][b][12], tid == f*BATCH+b
    T[0] = ux; T[1] = wx; T[2] = nx;
    T[3] = uy; T[4] = wy; T[5] = ny;
    T[6] = uz; T[7] = wz; T[8] = nz;
    T[9] = cx; T[10] = cy; T[11] = cz;
}

// ---------------------------------------------------------------------------
// Stage B1: per-(group,batch) intra-group prefix scan (in place, identity at
// group start) + emit the group-total affine.
// ---------------------------------------------------------------------------
__global__ void group_scan_kernel(float* __restrict__ tw,
                                  float* __restrict__ gw) {
    int tid = blockIdx.x * blockDim.x + threadIdx.x;   // NGROUP*BATCH, exact
    int b = tid % BATCH;
    int g = tid / BATCH;

    float M[9] = {1, 0, 0, 0, 1, 0, 0, 0, 1};
    float t[3] = {0, 0, 0};
    for (int k = 0; k < GROUP; ++k) {
        int f = g * GROUP + k;
        float* T = tw + ((size_t)f * BATCH + b) * 12;
        float Q[9], q[3];
#pragma unroll
        for (int j = 0; j < 9; ++j) Q[j] = T[j];
        q[0] = T[9]; q[1] = T[10]; q[2] = T[11];
#pragma unroll
        for (int j = 0; j < 9; ++j) T[j] = M[j];
        T[9] = t[0]; T[10] = t[1]; T[11] = t[2];
        compose(M, t, Q, q);
    }
    float* G = gw + (size_t)tid * 12;   // [g][b][12]
#pragma unroll
    for (int j = 0; j < 9; ++j) G[j] = M[j];
    G[9] = t[0]; G[10] = t[1]; G[11] = t[2];
}

// ---------------------------------------------------------------------------
// Stage B2: per-batch prefix scan over the 24 group totals (in place).
// ---------------------------------------------------------------------------
__global__ void group_prefix_kernel(float* __restrict__ gw) {
    int b = blockIdx.x * blockDim.x + threadIdx.x;     // exact grid (512)
    float M[9] = {1, 0, 0, 0, 1, 0, 0, 0, 1};
    float t[3] = {0, 0, 0};
    for (int g = 0; g < NGROUP; ++g) {
        float* G = gw + ((size_t)g * BATCH + b) * 12;
        float Q[9], q[3];
#pragma unroll
        for (int j = 0; j < 9; ++j) Q[j] = G[j];
        q[0] = G[9]; q[1] = G[10]; q[2] = G[11];
#pragma unroll
        for (int j = 0; j < 9; ++j) G[j] = M[j];
        G[9] = t[0]; G[10] = t[1]; G[11] = t[2];
        compose(M, t, Q, q);
    }
}

// ---------------------------------------------------------------------------
// Stage B3: fold group prefix into each fragment's local prefix (parallel):
// tw[f] <- gw[f/GROUP] o tw[f].
// ---------------------------------------------------------------------------
__global__ void combine_prefix_kernel(float* __restrict__ tw,
                                      const float* __restrict__ gw) {
    int tid = blockIdx.x * blockDim.x + threadIdx.x;   // NFRAG*BATCH, exact
    int b = tid % BATCH;
    int f = tid / BATCH;
    int g = f / GROUP;

    const float* G = gw + ((size_t)g * BATCH + b) * 12;
    float M[9], t[3];
#pragma unroll
    for (int j = 0; j < 9; ++j) M[j] = G[j];
    t[0] = G[9]; t[1] = G[10]; t[2] = G[11];

    float* T = tw + (size_t)tid * 12;
    float Q[9], q[3];
#pragma unroll
    for (int j = 0; j < 9; ++j) Q[j] = T[j];
    q[0] = T[9]; q[1] = T[10]; q[2] = T[11];

    compose(M, t, Q, q);
#pragma unroll
    for (int j = 0; j < 9; ++j) T[j] = M[j];
    T[9] = t[0]; T[10] = t[1]; T[11] = t[2];
}

// ---------------------------------------------------------------------------
// Stage C: apply global prefix affine to the 16 local coords of each
// (fragment,batch) pair with one V_WMMA_F32_16X16X4_F32 per wave:
//   D(16x16) = A(16x4) x B(4x16), A rows 0..2 = [M | t], B cols = (x,y,z,1).
// D: lane n<16, VGPRs 0..2 = out x,y,z of point n.
// ---------------------------------------------------------------------------
__global__ void wmma_apply_kernel(float* __restrict__ out,
                                  const float* __restrict__ tw) {
    int tid    = blockIdx.x * blockDim.x + threadIdx.x;
    int waveId = tid >> 5;                 // wave-uniform
    int lane   = threadIdx.x & 31;
    int b = waveId % BATCH;
    int f = waveId / BATCH;

    int  r  = lane & 15;
    bool hi = lane >= 16;

    // A operand: [M | t]; branchless clamped gather
    const float* T = tw + ((size_t)f * BATCH + b) * 12;
    int i0 = (r < 3) ? (3 * r + (hi ? 2 : 0)) : 0;
    int i1 = (r < 3) ? (hi ? (9 + r) : (3 * r + 1)) : 0;
    float a0 = T[i0];
    float a1 = T[i1];
    a0 = (r < 3) ? a0 : 0.0f;
    a1 = (r < 3) ? a1 : 0.0f;

    // B operand: point r of this fragment; lanes<16 carry (x,y), lanes>=16 (z,1)
    size_t n = (size_t)f * FRAG_S + r;
    float* cp = out + (n * BATCH + b) * 3;
    float x = cp[0], y = cp[1], z = cp[2];

    v2f Av = {a0, a1};
    v2f Bv = {hi ? z : x, hi ? 1.0f : y};
    v8f acc = {};
    // 8 args: (neg_a, A, neg_b, B, c_mod, C, reuse_a, reuse_b)
    acc = __builtin_amdgcn_wmma_f32_16x16x4_f32(
        false, Av, false, Bv, (short)0, acc, false, false);

    if (!hi) {                             // rows 0..2 live in lanes 0..15
        cp[0] = acc[0];
        cp[1] = acc[1];
        cp[2] = acc[2];
    }
}

extern "C" void kernel_launch(void* const* d_in, const int* in_sizes, int n_in,
                              void* d_out, int out_size, void* d_ws, size_t ws_size,
                              hipStream_t stream) {
    (void)in_sizes; (void)n_in; (void)out_size; (void)ws_size;
    const float* dih = (const float*)d_in[0];
    float* out = (float*)d_out;
    float* tw  = (float*)d_ws;                               // NFRAG*BATCH*12 f32 (~9 MB)
    float* gw  = tw + (size_t)NFRAG * BATCH * 12;            // NGROUP*BATCH*12 f32 (~0.6 MB)

    {   // Stage A: 196608 threads, 16 serial steps each
        dim3 grid(NFRAG * BATCH / 256), block(256);
        nerf_fragment_kernel<<<grid, block, 0, stream>>>(dih, out, tw);
    }
    {   // Stage B1: 12288 threads, 16 serial compositions each
        dim3 grid(NGROUP * BATCH / 256), block(256);
        group_scan_kernel<<<grid, block, 0, stream>>>(tw, gw);
    }
    {   // Stage B2: 512 threads, 24 serial compositions each
        dim3 grid(BATCH / 256), block(256);
        group_prefix_kernel<<<grid, block, 0, stream>>>(gw);
    }
    {   // Stage B3: 196608 threads, 1 composition each (fully parallel)
        dim3 grid(NFRAG * BATCH / 256), block(256);
        combine_prefix_kernel<<<grid, block, 0, stream>>>(tw, gw);
    }
    {   // Stage C: one wave per (fragment,batch), exact grid (EXEC all-1s)
        dim3 grid((size_t)NFRAG * BATCH * 32 / 256), block(256);
        wmma_apply_kernel<<<grid, block, 0, stream>>>(out, tw);
    }
}